// GCNLayer_12635793785680
// MI455X (gfx1250) — compile-verified
//
#include <hip/hip_runtime.h>
#include <hip/hip_bf16.h>

#define IN_F  256
#define OUT_F 128

typedef __attribute__((ext_vector_type(2))) float v2f;
typedef __attribute__((ext_vector_type(8))) float v8f;

// ---------------------------------------------------------------------------
// Kernel 1: h = x @ W + b via V_WMMA_F32_16X16X4_F32 (full f32 precision).
// One wave (32 lanes) computes one 16x16 tile of h. Block = 256 thr = 8 waves
// -> covers 16 rows x 128 cols (= all OUT_F) per block.
// A (16x4 f32) layout: lanes 0-15 = rows M=0..15, VGPR0 = K(=k0+2*half),
// VGPR1 = K+1. B (4x16) symmetric with N across lanes. C/D: VGPR v holds
// element (M = v + 8*half, N = lane%16).
// ---------------------------------------------------------------------------
__global__ void __launch_bounds__(256)
gcn_gemm_wmma(const float* __restrict__ x, const float* __restrict__ w,
              const float* __restrict__ bias, float* __restrict__ h,
              int n_nodes) {
    const int lane = threadIdx.x & 31;
    const int wave = threadIdx.x >> 5;     // 0..7 -> column tile
    const int half = lane >> 4;            // 0 or 1
    const int l16  = lane & 15;

    const int row_base = blockIdx.x * 16;
    const int col_base = wave * 16;

    // Clamp instead of predicating: WMMA requires EXEC all-1s.
    int m = row_base + l16;
    if (m >= n_nodes) m = n_nodes - 1;
    const int n = col_base + l16;

    const float* xrow = x + (size_t)m * IN_F;
    const float* wcol = w + n;

    v8f acc = {};
#pragma unroll 8
    for (int k0 = 0; k0 < IN_F; k0 += 4) {
        const int ka = k0 + 2 * half;
        v2f a = *(const v2f*)(xrow + ka);               // K=ka, ka+1 for row m
        v2f b;
        b.x = wcol[(size_t)ka * OUT_F];                 // K=ka,   col n
        b.y = wcol[(size_t)(ka + 1) * OUT_F];           // K=ka+1, col n
        acc = __builtin_amdgcn_wmma_f32_16x16x4_f32(
            /*neg_a=*/false, a, /*neg_b=*/false, b,
            /*c_mod=*/(short)0, acc, /*reuse_a=*/false, /*reuse_b=*/false);
    }

    const float bn = bias[n];
    float* hp = h + (size_t)(row_base + 8 * half) * OUT_F + n;

    if (row_base + 16 <= n_nodes) {
        // Fast path: full tile, straight-line coalesced stores (always taken
        // when n_nodes % 16 == 0, e.g. 100000).
#pragma unroll
        for (int v = 0; v < 8; ++v)
            hp[(size_t)v * OUT_F] = acc[v] + bn;
    } else {
        const int r0 = row_base + 8 * half;
#pragma unroll
        for (int v = 0; v < 8; ++v)
            if (r0 + v < n_nodes) hp[(size_t)v * OUT_F] = acc[v] + bn;
    }
}

// ---------------------------------------------------------------------------
// Kernel 2: zero the output (harness poisons it with 0xAA).
// ---------------------------------------------------------------------------
__global__ void gcn_zero(float4* __restrict__ out, long long n4) {
    long long i = (long long)blockIdx.x * blockDim.x + threadIdx.x;
    if (i < n4) out[i] = make_float4(0.f, 0.f, 0.f, 0.f);
}

// ---------------------------------------------------------------------------
// Kernel 3: per-edge gather+scale+scatter-add.
// One wave per edge: lane L loads h[col][4L..4L+3] (float4, 512B coalesced
// per wave, L2-resident since h = 51.2MB < 192MB L2) and issues 4
// global_atomic_add_f32 into out[row].
// ---------------------------------------------------------------------------
__global__ void __launch_bounds__(256)
gcn_scatter(const float* __restrict__ h, const int* __restrict__ erow,
            const int* __restrict__ ecol, const float* __restrict__ eval,
            float* __restrict__ out, int n_edges) {
    const int edge = blockIdx.x * 8 + (int)(threadIdx.x >> 5);  // wave-uniform
    const int lane = threadIdx.x & 31;
    if (edge >= n_edges) return;

    const int   r = erow[edge];
    const int   c = ecol[edge];
    const float v = eval[edge];

    const float4 hm = *(const float4*)(h + (size_t)c * OUT_F + lane * 4);
    float* op = out + (size_t)r * OUT_F + lane * 4;
    atomicAdd(op + 0, hm.x * v);
    atomicAdd(op + 1, hm.y * v);
    atomicAdd(op + 2, hm.z * v);
    atomicAdd(op + 3, hm.w * v);
}

extern "C" void kernel_launch(void* const* d_in, const int* in_sizes, int n_in,
                              void* d_out, int out_size, void* d_ws, size_t ws_size,
                              hipStream_t stream) {
    const float* x    = (const float*)d_in[0];
    const int*   erow = (const int*)  d_in[1];
    const int*   ecol = (const int*)  d_in[2];
    const float* ev   = (const float*)d_in[3];
    const float* w    = (const float*)d_in[4];
    const float* bias = (const float*)d_in[5];
    float*       out  = (float*)d_out;
    float*       h    = (float*)d_ws;      // [n_nodes, OUT_F] scratch, 51.2 MB

    const int n_nodes = in_sizes[0] / IN_F;
    const int n_edges = in_sizes[1];

    // 1) zero output
    const long long n4 = (long long)n_nodes * OUT_F / 4;
    gcn_zero<<<(unsigned)((n4 + 255) / 256), 256, 0, stream>>>((float4*)out, n4);

    // 2) dense transform h = x@W + b (WMMA f32)
    const int row_tiles = (n_nodes + 15) / 16;   // 6250 for N=100000
    gcn_gemm_wmma<<<row_tiles, 256, 0, stream>>>(x, w, bias, h, n_nodes);

    // 3) sparse aggregate: one wave per edge, 8 edges per 256-thread block
    const int blocks = (n_edges + 7) / 8;
    gcn_scatter<<<blocks, 256, 0, stream>>>(h, erow, ecol, ev, out, n_edges);
}